// SetAbstraction_14559939133620
// MI455X (gfx1250) — compile-verified
//
#include <hip/hip_runtime.h>
#include <math.h>

typedef float v2f __attribute__((ext_vector_type(2)));
typedef float v8f __attribute__((ext_vector_type(8)));

#define B_       16
#define NPTS     4096
#define NCENT    1024
#define NSAMP    32
#define NROWS    (B_ * NCENT * NSAMP)      // 524288
#define RAD2     0.16f
#define MASK_SCALE 1e11f
#define BN_EPS   1e-3f

// ---------------------------------------------------------------------------
// 0) zero the BN accumulator buffers (ws is poisoned with 0xAA)
// ---------------------------------------------------------------------------
__global__ void zero_stats_kernel(float* statsSum, float* statsSq) {
    int t = threadIdx.x;            // 512 threads
    if (t < 256) statsSum[t] = 0.f;
    else         statsSq[t - 256] = 0.f;
}

// ---------------------------------------------------------------------------
// 1) Farthest point sampling: one block per batch, mask+dist resident in LDS
// ---------------------------------------------------------------------------
__global__ void fps_kernel(const float* __restrict__ xyz, int* __restrict__ centIdx) {
    __shared__ float mask[NPTS];
    __shared__ float dist[NPTS];
    __shared__ float redV[256];
    __shared__ int   redI[256];
    __shared__ int   s_last;
    __shared__ float s_px, s_py, s_pz;

    const int b   = blockIdx.x;
    const int tid = threadIdx.x;
    const float* X = xyz + (long)b * NPTS * 3;

    for (int i = tid; i < NPTS; i += 256) mask[i] = 1.f;
    if (tid == 0) { s_last = 0; centIdx[b * NCENT] = 0; }
    __syncthreads();

    for (int t = 1; t < NCENT; ++t) {
        if (tid == 0) {
            int li = s_last;
            s_px = X[li * 3 + 0]; s_py = X[li * 3 + 1]; s_pz = X[li * 3 + 2];
        }
        __syncthreads();
        float px = s_px, py = s_py, pz = s_pz;
        float bv = -1.f; int bi = 0;
        for (int i = tid; i < NPTS; i += 256) {
            float dx = X[i * 3 + 0] - px;
            float dy = X[i * 3 + 1] - py;
            float dz = X[i * 3 + 2] - pz;
            float nd = sqrtf(dx * dx + dy * dy + dz * dz);
            float d  = nd * mask[i];
            dist[i] = d;
            if (d > bv) { bv = d; bi = i; }   // first-max within this thread's stride
        }
        redV[tid] = bv; redI[tid] = bi;
        __syncthreads();
        for (int s = 128; s > 0; s >>= 1) {   // argmax, ties -> smaller index
            if (tid < s) {
                float ov = redV[tid + s]; int oi = redI[tid + s];
                if (ov > redV[tid] || (ov == redV[tid] && oi < redI[tid])) {
                    redV[tid] = ov; redI[tid] = oi;
                }
            }
            __syncthreads();
        }
        if (tid == 0) { s_last = redI[0]; centIdx[b * NCENT + t] = redI[0]; }
        for (int i = tid; i < NPTS; i += 256) {
            float mk = mask[i];
            mask[i] = fminf(dist[i] * mk * MASK_SCALE, mk);
        }
        __syncthreads();
    }
}

// ---------------------------------------------------------------------------
// 2) Ball query: stable top-32 of min(||x-c||, r^2) via insertion (matches
//    JAX stable argsort tie-breaking). One thread per centroid; point tiles
//    staged in LDS. All centroids in a block share one batch.
// ---------------------------------------------------------------------------
#define QTILE 128
__global__ void ballquery_kernel(const float* __restrict__ xyz,
                                 const int* __restrict__ centIdx,
                                 int* __restrict__ gidx) {
    __shared__ float sp[QTILE * 3];
    const int tid = threadIdx.x;                       // 128
    const long gc = (long)blockIdx.x * 128 + tid;      // global centroid
    const int b   = (int)(gc >> 10);                   // / NCENT
    const float* X = xyz + (long)b * NPTS * 3;
    const int ci = centIdx[gc];
    const float cx = X[ci * 3 + 0], cy = X[ci * 3 + 1], cz = X[ci * 3 + 2];

    float td[NSAMP]; int ti[NSAMP];
    for (int j = 0; j < NSAMP; ++j) { td[j] = 3.4e38f; ti[j] = 0; }

    for (int p0 = 0; p0 < NPTS; p0 += QTILE) {
        __syncthreads();
        for (int f = tid; f < QTILE * 3; f += 128) sp[f] = X[p0 * 3 + f];
        __syncthreads();
        for (int q = 0; q < QTILE; ++q) {
            float dx = sp[q * 3 + 0] - cx;
            float dy = sp[q * 3 + 1] - cy;
            float dz = sp[q * 3 + 2] - cz;
            float d  = fminf(sqrtf(dx * dx + dy * dy + dz * dz), RAD2);
            if (d < td[NSAMP - 1]) {           // equal -> later index dropped (stable)
                int j = NSAMP - 1;
                while (j > 0 && td[j - 1] > d) {   // strict: equals keep order
                    td[j] = td[j - 1]; ti[j] = ti[j - 1]; --j;
                }
                td[j] = d; ti[j] = p0 + q;
            }
        }
    }
    for (int j = 0; j < NSAMP; ++j) gidx[gc * NSAMP + j] = ti[j];
}

// ---------------------------------------------------------------------------
// 3) Layer 1 (K=3, not worth WMMA): gather + x@W1 + b1, online BN stats
//    512 threads = 8 rows x 64 channels
// ---------------------------------------------------------------------------
__global__ void mlp1_kernel(const float* __restrict__ xyz, const int* __restrict__ gidx,
                            const float* __restrict__ W1, const float* __restrict__ b1,
                            float* __restrict__ h1,
                            float* __restrict__ statsSum, float* __restrict__ statsSq) {
    __shared__ float sW[3 * 64];
    __shared__ float sB[64];
    __shared__ float sSum[64], sSq[64];
    const int tid = threadIdx.x;
    if (tid < 192) sW[tid] = W1[tid];
    if (tid < 64) { sB[tid] = b1[tid]; sSum[tid] = 0.f; sSq[tid] = 0.f; }
    __syncthreads();

    const int  c   = tid & 63;
    const long row = (long)blockIdx.x * 8 + (tid >> 6);
    const int  b   = (int)(row >> 15);                 // / (NCENT*NSAMP)
    const int  idx = gidx[row];
    const float* p = xyz + ((long)b * NPTS + idx) * 3;
    const float x = p[0], y = p[1], z = p[2];
    const float v = x * sW[c] + y * sW[64 + c] + z * sW[128 + c] + sB[c];
    h1[row * 64 + c] = v;
    atomicAdd(&sSum[c], v);
    atomicAdd(&sSq[c], v * v);
    __syncthreads();
    if (tid < 64) { atomicAdd(&statsSum[tid], sSum[tid]); atomicAdd(&statsSq[tid], sSq[tid]); }
}

// ---------------------------------------------------------------------------
// 4) BN finalize: fold mean/var/gamma/beta into per-channel affine (scale,shift)
// ---------------------------------------------------------------------------
__global__ void bnfin_kernel(const float* __restrict__ sum, const float* __restrict__ sq,
                             const float* __restrict__ g, const float* __restrict__ bt,
                             float* __restrict__ sc, float* __restrict__ sh) {
    const int t = threadIdx.x;
    const float inv = 1.f / (float)NROWS;
    const float mean = sum[t] * inv;
    const float var  = sq[t] * inv - mean * mean;
    const float s = rsqrtf(var + BN_EPS) * g[t];
    sc[t] = s;
    sh[t] = bt[t] - mean * s;
}

// ---------------------------------------------------------------------------
// 5) WMMA GEMM (K=64, N templated 64/128), BN(prev)+ReLU fused on the A load,
//    online BN stats on the output. 128 threads = 4 waves, each wave owns a
//    16-row stripe; V_WMMA_F32_16X16X4_F32, 16 steps over K.
//
//    W is pre-swizzled in LDS into per-lane fragment order (float2 per lane
//    per K-step) so every B fragment is ONE ds_load_b64 at an immediate
//    offset; A fragment pairs are contiguous already. All 16 A and 16 B
//    fragments are register-resident before the WMMA chain issues.
//
//    STORE_H: write pre-BN activations (layers 2).
//    MINMAX:  emit per-16-row-tile max/min of pre-BN values (layer 3) --
//             relu(s*x+t) is monotone in x per channel, so the sample-max
//             only needs max/min of x; the 268MB h3 tensor is never stored.
// ---------------------------------------------------------------------------
template <int N, bool STORE_H, bool MINMAX>
__global__ void gemm_bn_wmma_kernel(const float* __restrict__ Ain,
                                    const float* __restrict__ scPrev,
                                    const float* __restrict__ shPrev,
                                    const float* __restrict__ W,
                                    const float* __restrict__ bias,
                                    float* __restrict__ Hout,
                                    float* __restrict__ maxP,
                                    float* __restrict__ minP,
                                    float* __restrict__ statsSum,
                                    float* __restrict__ statsSq) {
    __shared__ v2f   sWf[32 * N];         // fragment-packed W: [chunk][ks][lane]
    __shared__ float sA[64 * 64];
    __shared__ float sBias[N];
    __shared__ float sScale[64], sShift[64];
    __shared__ float sSum[N], sSq[N];

    const int tid = threadIdx.x;          // 128

    // pack W into per-lane fragment order: idx = ((c*16 + ks)*32 + h*16 + m)
    for (int idx = tid; idx < 32 * N; idx += 128) {
        const int m  = idx & 15;
        const int h  = (idx >> 4) & 1;
        const int ks = (idx >> 5) & 15;
        const int c  = idx >> 9;
        const int kk  = ks * 4 + h * 2;
        const int col = c * 16 + m;
        v2f w;
        w.x = W[kk * N + col];
        w.y = W[(kk + 1) * N + col];
        sWf[idx] = w;
    }
    for (int f = tid; f < N; f += 128) { sBias[f] = bias[f]; sSum[f] = 0.f; sSq[f] = 0.f; }
    if (tid < 64) { sScale[tid] = scPrev[tid]; sShift[tid] = shPrev[tid]; }
    __syncthreads();

    const long rowBase0 = (long)blockIdx.x * 64;
    // stage A tile with fused BN-affine + ReLU of the previous layer
    for (int f = tid; f < 64 * 64; f += 128) {
        const int r = f >> 6, k = f & 63;
        float a = Ain[(rowBase0 + r) * 64 + k];
        a = a * sScale[k] + sShift[k];
        sA[f] = fmaxf(a, 0.f);
    }
    __syncthreads();

    const int wave = tid >> 5, lane = tid & 31;
    const int half = lane >> 4, m = lane & 15;
    const int rtile = wave * 16;

    // register-resident A fragments (pairs are contiguous in sA rows)
    v2f afrag[16];
    {
        const v2f* pA = (const v2f*)(sA + (rtile + m) * 64);
        #pragma unroll
        for (int ks = 0; ks < 16; ++ks) afrag[ks] = pA[ks * 2 + half];
    }

    #pragma unroll
    for (int nc = 0; nc < N / 16; ++nc) {
        const int n0  = nc * 16;
        const int col = n0 + m;

        // register-resident B fragments: one b64 LDS load each, imm offsets
        v2f bfrag[16];
        {
            const v2f* pB = sWf + (nc * 512 + lane);
            #pragma unroll
            for (int ks = 0; ks < 16; ++ks) bfrag[ks] = pB[ks * 32];
        }

        v8f acc;
        const float bval = sBias[col];            // each lane owns one column
        #pragma unroll
        for (int r = 0; r < 8; ++r) acc[r] = bval;

        #pragma unroll
        for (int ks = 0; ks < 16; ++ks)
            acc = __builtin_amdgcn_wmma_f32_16x16x4_f32(
                false, afrag[ks], false, bfrag[ks], (short)0, acc, false, false);

        float psum = 0.f, psq = 0.f, pmax = -3.4e38f, pmin = 3.4e38f;
        #pragma unroll
        for (int r = 0; r < 8; ++r) {             // VGPR r -> row r (half 0) / r+8 (half 1)
            const float v = acc[r];
            psum += v; psq += v * v;
            pmax = fmaxf(pmax, v); pmin = fminf(pmin, v);
            if (STORE_H) {
                const long row = rowBase0 + rtile + half * 8 + r;
                Hout[row * N + col] = v;
            }
        }
        atomicAdd(&sSum[col], psum);
        atomicAdd(&sSq[col],  psq);
        if (MINMAX) {                             // combine row-halves across lane^16
            const float omx = __shfl_xor(pmax, 16);
            const float omn = __shfl_xor(pmin, 16);
            pmax = fmaxf(pmax, omx); pmin = fminf(pmin, omn);
            if (half == 0) {
                const long t = (rowBase0 + rtile) >> 4;   // 16-row tile id
                maxP[t * N + col] = pmax;
                minP[t * N + col] = pmin;
            }
        }
    }
    __syncthreads();
    for (int f = tid; f < N; f += 128) {
        atomicAdd(&statsSum[f], sSum[f]);
        atomicAdd(&statsSq[f],  sSq[f]);
    }
}

// ---------------------------------------------------------------------------
// 6) Final: out[bc,c] = max over 32 samples of relu(s3*x+t3) using monotonicity
// ---------------------------------------------------------------------------
__global__ void finalmax_kernel(const float* __restrict__ maxP, const float* __restrict__ minP,
                                const float* __restrict__ sc3, const float* __restrict__ sh3,
                                float* __restrict__ out) {
    const int  c  = threadIdx.x;                  // 128
    const long bc = blockIdx.x;                   // 16384 centroids total
    const float xmax = fmaxf(maxP[(bc * 2) * 128 + c], maxP[(bc * 2 + 1) * 128 + c]);
    const float xmin = fminf(minP[(bc * 2) * 128 + c], minP[(bc * 2 + 1) * 128 + c]);
    const float s = sc3[c], t = sh3[c];
    const float v = (s >= 0.f) ? (s * xmax + t) : (s * xmin + t);
    out[bc * 128 + c] = fmaxf(v, 0.f);
}

// ---------------------------------------------------------------------------
extern "C" void kernel_launch(void* const* d_in, const int* in_sizes, int n_in,
                              void* d_out, int out_size, void* d_ws, size_t ws_size,
                              hipStream_t stream) {
    const float* xyz = (const float*)d_in[0];
    const float* W1  = (const float*)d_in[1];
    const float* b1  = (const float*)d_in[2];
    const float* g1  = (const float*)d_in[3];
    const float* bt1 = (const float*)d_in[4];
    const float* W2  = (const float*)d_in[5];
    const float* b2  = (const float*)d_in[6];
    const float* g2  = (const float*)d_in[7];
    const float* bt2 = (const float*)d_in[8];
    const float* W3  = (const float*)d_in[9];
    const float* b3  = (const float*)d_in[10];
    const float* g3  = (const float*)d_in[11];
    const float* bt3 = (const float*)d_in[12];
    float* out = (float*)d_out;

    // workspace layout (256B-aligned)
    char* ws = (char*)d_ws;
    size_t off = 0;
    auto take = [&](size_t bytes) -> char* {
        char* p = ws + off;
        off = (off + bytes + 255) & ~(size_t)255;
        return p;
    };
    int*   centIdx = (int*)  take((size_t)B_ * NCENT * 4);
    int*   gidx    = (int*)  take((size_t)NROWS * 4);
    float* h1      = (float*)take((size_t)NROWS * 64 * 4);
    float* h2      = (float*)take((size_t)NROWS * 64 * 4);
    float* maxP    = (float*)take((size_t)(NROWS / 16) * 128 * 4);
    float* minP    = (float*)take((size_t)(NROWS / 16) * 128 * 4);
    float* statsSum = (float*)take(256 * 4);   // ch: [0,64)=L1 [64,128)=L2 [128,256)=L3
    float* statsSq  = (float*)take(256 * 4);
    float* bnScale  = (float*)take(256 * 4);
    float* bnShift  = (float*)take(256 * 4);

    zero_stats_kernel<<<1, 512, 0, stream>>>(statsSum, statsSq);
    fps_kernel<<<B_, 256, 0, stream>>>(xyz, centIdx);
    ballquery_kernel<<<(B_ * NCENT) / 128, 128, 0, stream>>>(xyz, centIdx, gidx);

    // layer 1 (K=3)
    mlp1_kernel<<<NROWS / 8, 512, 0, stream>>>(xyz, gidx, W1, b1, h1,
                                               statsSum + 0, statsSq + 0);
    bnfin_kernel<<<1, 64, 0, stream>>>(statsSum + 0, statsSq + 0, g1, bt1,
                                       bnScale + 0, bnShift + 0);

    // layer 2 (64 -> 64), WMMA, BN1+ReLU fused on load
    gemm_bn_wmma_kernel<64, true, false><<<NROWS / 64, 128, 0, stream>>>(
        h1, bnScale + 0, bnShift + 0, W2, b2,
        h2, nullptr, nullptr, statsSum + 64, statsSq + 64);
    bnfin_kernel<<<1, 64, 0, stream>>>(statsSum + 64, statsSq + 64, g2, bt2,
                                       bnScale + 64, bnShift + 64);

    // layer 3 (64 -> 128), WMMA, BN2+ReLU fused on load; emits tile max/min only
    gemm_bn_wmma_kernel<128, false, true><<<NROWS / 64, 128, 0, stream>>>(
        h2, bnScale + 64, bnShift + 64, W3, b3,
        nullptr, maxP, minP, statsSum + 128, statsSq + 128);
    bnfin_kernel<<<1, 128, 0, stream>>>(statsSum + 128, statsSq + 128, g3, bt3,
                                        bnScale + 128, bnShift + 128);

    // BN3 + ReLU + max over the 32 samples (monotone per channel)
    finalmax_kernel<<<B_ * NCENT, 128, 0, stream>>>(maxP, minP,
                                                    bnScale + 128, bnShift + 128, out);
}